// TrafficSignalController_44049184588392
// MI455X (gfx1250) — compile-verified
//
#include <hip/hip_runtime.h>

#define IN_DIM  64
#define HID_DIM 128
#define OUT_DIM 2

typedef __attribute__((ext_vector_type(2))) float v2f;
typedef __attribute__((ext_vector_type(8))) float v8f;

// ---------------------------------------------------------------------------
// Init: agg = 0, deg = 1 (self-loop contribution to in-degree)
// ---------------------------------------------------------------------------
__global__ void gcn_init_kernel(float* __restrict__ deg, float* __restrict__ agg,
                                int N) {
  long long idx = (long long)blockIdx.x * blockDim.x + threadIdx.x;
  long long total = (long long)N * HID_DIM;
  if (idx < total) agg[idx] = 0.0f;
  if (idx < N) deg[idx] = 1.0f;
}

// ---------------------------------------------------------------------------
// Degree: deg[dst[e]] += 1   (3.2M scalar fp32 atomics, L2 resident)
// ---------------------------------------------------------------------------
__global__ void gcn_degree_kernel(const int* __restrict__ dst,
                                  float* __restrict__ deg, int E) {
  int e = blockIdx.x * blockDim.x + threadIdx.x;
  if (e < E) unsafeAtomicAdd(&deg[dst[e]], 1.0f);
}

// ---------------------------------------------------------------------------
// dinv = deg > 0 ? rsqrt(deg) : 0   (in place)
// ---------------------------------------------------------------------------
__global__ void gcn_dinv_kernel(float* __restrict__ deg, int N) {
  int i = blockIdx.x * blockDim.x + threadIdx.x;
  if (i < N) {
    float d = deg[i];
    deg[i] = (d > 0.0f) ? rsqrtf(d) : 0.0f;
  }
}

// ---------------------------------------------------------------------------
// h = x @ W_gcn via V_WMMA_F32_16X16X4_F32 (exact fp32 matrix path).
// Block = 256 threads = 8 waves (wave32). Block -> 16-row M strip.
// Wave w -> N tile [16w, 16w+16). K loop: 16 WMMA steps of K=4.
// W_gcn (64x128 = 32KB) staged in LDS (320KB/WGP budget).
// A layout (16x4 f32): lane l holds M=l%16, K = vgpr + 2*(l/16)
// B layout (4x16 f32): lane l holds N=l%16, K = vgpr + 2*(l/16)
// D layout (16x16 f32): vgpr v, lane l -> M = v + 8*(l/16), N = l%16
// ---------------------------------------------------------------------------
__global__ __launch_bounds__(256) void gcn_gemm_wmma_kernel(
    const float* __restrict__ X, const float* __restrict__ W,
    float* __restrict__ H, int N) {
  __shared__ float Wl[IN_DIM * HID_DIM];
  for (int i = threadIdx.x; i < IN_DIM * HID_DIM; i += 256) Wl[i] = W[i];
  __syncthreads();

  const int wave  = threadIdx.x >> 5;
  const int lane  = threadIdx.x & 31;
  const int half  = lane >> 4;     // 0 or 1
  const int lm    = lane & 15;
  const int mbase = blockIdx.x * 16;
  const int nbase = wave * 16;

  int arow = mbase + lm;
  if (arow >= N) arow = N - 1;     // tail safety; EXEC stays all-ones for WMMA
  const float* xrow = X + (long long)arow * IN_DIM;

  v8f acc = {};
#pragma unroll
  for (int k0 = 0; k0 < IN_DIM; k0 += 4) {
    const int ka = k0 + 2 * half;
    v2f a, b;
    a.x = xrow[ka];
    a.y = xrow[ka + 1];
    b.x = Wl[ka * HID_DIM + nbase + lm];
    b.y = Wl[(ka + 1) * HID_DIM + nbase + lm];
    acc = __builtin_amdgcn_wmma_f32_16x16x4_f32(
        /*neg_a=*/false, a, /*neg_b=*/false, b,
        /*c_mod=*/(short)0, acc, /*reuse_a=*/false, /*reuse_b=*/false);
  }

  float* hbase = H + (long long)(mbase + 8 * half) * HID_DIM + nbase + lm;
  if (mbase + 16 <= N) {
    // Full tile (the common case: N % 16 == 0): unconditional stores,
    // no per-element EXEC churn.
#pragma unroll
    for (int v = 0; v < 8; ++v) hbase[(long long)v * HID_DIM] = acc[v];
  } else {
#pragma unroll
    for (int v = 0; v < 8; ++v) {
      int m = mbase + v + 8 * half;
      if (m < N) H[(long long)m * HID_DIM + nbase + lm] = acc[v];
    }
  }
}

// ---------------------------------------------------------------------------
// Edge scatter: agg[dst] += h[src] * (dinv[src]*dinv[dst])
// ONE WAVE PER EDGE: edge id is wave-uniform -> src/dst/dinv scalarize to
// SMEM loads; each lane owns 4 channels (32 lanes x float4 = 128 ch), so the
// h[src] row is a single coalesced 512B wave read. 4 fp32 atomics per lane.
// All row traffic is L2-resident (h, agg each 51MB < 192MB L2).
// ---------------------------------------------------------------------------
__global__ __launch_bounds__(256) void gcn_scatter_kernel(
    const int* __restrict__ src, const int* __restrict__ dst,
    const float* __restrict__ dinv, const float* __restrict__ H,
    float* __restrict__ agg, int E) {
  int e = blockIdx.x * 8 + (threadIdx.x >> 5);   // 8 waves / block
  if (e >= E) return;
  int s = src[e];                                 // wave-uniform -> s_load
  int d = dst[e];
  float norm = dinv[s] * dinv[d];
  int c = (threadIdx.x & 31) * 4;
  float4 v = *(const float4*)(H + (long long)s * HID_DIM + c);
  float* ap = agg + (long long)d * HID_DIM + c;
  unsafeAtomicAdd(ap + 0, v.x * norm);
  unsafeAtomicAdd(ap + 1, v.y * norm);
  unsafeAtomicAdd(ap + 2, v.z * norm);
  unsafeAtomicAdd(ap + 3, v.w * norm);
}

// ---------------------------------------------------------------------------
// Finalize: out[i] = relu(agg[i] + h[i]*dinv[i]^2 + b_gcn) @ W_lin + b_lin
// ONE WAVE PER NODE: float4 per lane (coalesced 512B row reads), then a
// 5-step shfl_xor wave32 reduction for the two output dot products.
// ---------------------------------------------------------------------------
__global__ __launch_bounds__(256) void gcn_finalize_kernel(
    const float* __restrict__ agg, const float* __restrict__ H,
    const float* __restrict__ dinv, const float* __restrict__ bg,
    const float* __restrict__ Wlin, const float* __restrict__ blin,
    float* __restrict__ out, int N) {
  int i = blockIdx.x * 8 + (threadIdx.x >> 5);   // 8 waves / block
  if (i >= N) return;
  int lane = threadIdx.x & 31;
  int c = lane * 4;

  float dii = dinv[i];                            // wave-uniform
  float self = dii * dii;

  float4 a  = *(const float4*)(agg + (long long)i * HID_DIM + c);
  float4 h4 = *(const float4*)(H + (long long)i * HID_DIM + c);
  float4 b4 = *(const float4*)(bg + c);

  float v0 = fmaxf(a.x + h4.x * self + b4.x, 0.0f);
  float v1 = fmaxf(a.y + h4.y * self + b4.y, 0.0f);
  float v2 = fmaxf(a.z + h4.z * self + b4.z, 0.0f);
  float v3 = fmaxf(a.w + h4.w * self + b4.w, 0.0f);

  // W_lin is [128,2] row-major: channels (c..c+3) x 2 cols = 2 float4 loads
  float4 w01 = *(const float4*)(Wlin + c * OUT_DIM);        // ch c, c+1
  float4 w23 = *(const float4*)(Wlin + (c + 2) * OUT_DIM);  // ch c+2, c+3

  float o0 = v0 * w01.x + v1 * w01.z + v2 * w23.x + v3 * w23.z;
  float o1 = v0 * w01.y + v1 * w01.w + v2 * w23.y + v3 * w23.w;

#pragma unroll
  for (int off = 16; off > 0; off >>= 1) {
    o0 += __shfl_xor(o0, off, 32);
    o1 += __shfl_xor(o1, off, 32);
  }
  if (lane == 0) {
    out[(long long)i * OUT_DIM + 0] = o0 + blin[0];
    out[(long long)i * OUT_DIM + 1] = o1 + blin[1];
  }
}

// ---------------------------------------------------------------------------
extern "C" void kernel_launch(void* const* d_in, const int* in_sizes, int n_in,
                              void* d_out, int out_size, void* d_ws, size_t ws_size,
                              hipStream_t stream) {
  const float* x    = (const float*)d_in[0];
  const int*   eidx = (const int*)d_in[1];   // [2, E] flat: row0=src, row1=dst
  const float* Wg   = (const float*)d_in[2];
  const float* bg   = (const float*)d_in[3];
  const float* Wlin = (const float*)d_in[4];
  const float* blin = (const float*)d_in[5];
  float* out = (float*)d_out;

  const int N = in_sizes[0] / IN_DIM;
  const int E = in_sizes[1] / 2;
  const int* src = eidx;
  const int* dst = eidx + E;

  // Workspace layout (all fp32, 16B-aligned regions):
  // [deg/dinv: N][h: N*128][agg: N*128]  ~= 103 MB
  float* ws = (float*)d_ws;
  size_t degN = ((size_t)N + 3) & ~(size_t)3;
  float* deg = ws;
  float* H   = ws + degN;
  float* agg = H + (size_t)N * HID_DIM;

  const int TB = 256;

  long long total = (long long)N * HID_DIM;
  gcn_init_kernel<<<(unsigned)((total + TB - 1) / TB), TB, 0, stream>>>(deg, agg, N);

  gcn_degree_kernel<<<(E + TB - 1) / TB, TB, 0, stream>>>(dst, deg, E);

  gcn_dinv_kernel<<<(N + TB - 1) / TB, TB, 0, stream>>>(deg, N);

  gcn_gemm_wmma_kernel<<<(N + 15) / 16, 256, 0, stream>>>(x, Wg, H, N);

  // one wave per edge, 8 waves per block
  gcn_scatter_kernel<<<(E + 7) / 8, TB, 0, stream>>>(src, dst, deg, H, agg, E);

  // one wave per node, 8 waves per block
  gcn_finalize_kernel<<<(N + 7) / 8, TB, 0, stream>>>(
      agg, H, deg, bg, Wlin, blin, out, N);
}